// EXGNN_85993835200539
// MI455X (gfx1250) — compile-verified
//
#include <hip/hip_runtime.h>
#include <hip/hip_bf16.h>
#include <float.h>

typedef __attribute__((ext_vector_type(2))) float v2f;
typedef __attribute__((ext_vector_type(8))) float v8f;

// ---------------------------------------------------------------------------
// CSR construction: degree count -> exclusive scan -> column fill
// ---------------------------------------------------------------------------
__global__ void __launch_bounds__(256) k_count_deg(const int* __restrict__ dst,
                                                   int* __restrict__ deg, int E) {
    int i = blockIdx.x * 256 + threadIdx.x;
    if (i < E) atomicAdd(&deg[dst[i]], 1);
}

// single-block exclusive scan (n <= ~50k, chunked Hillis-Steele in LDS)
__global__ void __launch_bounds__(1024) k_scan(const int* __restrict__ deg,
                                               int* __restrict__ rowptr,
                                               int* __restrict__ cursor, int n) {
    __shared__ int buf[1024];
    const int t = threadIdx.x;
    int carry = 0;
    for (int base = 0; base < n; base += 1024) {
        int v = (base + t < n) ? deg[base + t] : 0;
        buf[t] = v;
        __syncthreads();
        for (int off = 1; off < 1024; off <<= 1) {
            int x = (t >= off) ? buf[t - off] : 0;
            __syncthreads();
            buf[t] += x;
            __syncthreads();
        }
        int incl = buf[t];
        int excl = incl - v;
        if (base + t < n) {
            rowptr[base + t] = carry + excl;
            cursor[base + t] = carry + excl;
        }
        int total = buf[1023];
        __syncthreads();
        carry += total;
    }
    if (t == 0) rowptr[n] = carry;
}

__global__ void __launch_bounds__(256) k_fill_col(const int* __restrict__ src,
                                                  const int* __restrict__ dst,
                                                  int* __restrict__ cursor,
                                                  int* __restrict__ col, int E) {
    int i = blockIdx.x * 256 + threadIdx.x;
    if (i < E) {
        int pos = atomicAdd(&cursor[dst[i]], 1);
        col[pos] = src[i];
    }
}

// ---------------------------------------------------------------------------
// Mean aggregation (gather form): one wave32 per destination node.
// DCH = number of 128-float chunks per row (1 -> D=128, 2 -> D=256).
// Edge loop unrolled x2 to keep two global_load_b128 gathers in flight.
// ---------------------------------------------------------------------------
template <int DCH>
__global__ void __launch_bounds__(256) k_gather_mean(const float* __restrict__ x,
                                                     const int* __restrict__ rowptr,
                                                     const int* __restrict__ deg,
                                                     const int* __restrict__ col,
                                                     float* __restrict__ out,
                                                     int n, int relu) {
    const int wave = threadIdx.x >> 5;
    const int lane = threadIdx.x & 31;
    const int node = blockIdx.x * 8 + wave;
    if (node >= n) return;
    const int DD = DCH * 128;
    const int start = rowptr[node];
    const int d = deg[node];
    float acc[DCH * 4];
#pragma unroll
    for (int i = 0; i < DCH * 4; ++i) acc[i] = 0.f;
    int e = 0;
    for (; e + 1 < d; e += 2) {
        int s0 = __builtin_amdgcn_readfirstlane(col[start + e]);
        int s1 = __builtin_amdgcn_readfirstlane(col[start + e + 1]);
        const float* xr0 = x + (size_t)s0 * DD;
        const float* xr1 = x + (size_t)s1 * DD;
#pragma unroll
        for (int c = 0; c < DCH; ++c) {
            float4 v0 = *(const float4*)(xr0 + c * 128 + lane * 4);
            float4 v1 = *(const float4*)(xr1 + c * 128 + lane * 4);
            acc[c * 4 + 0] += v0.x + v1.x;
            acc[c * 4 + 1] += v0.y + v1.y;
            acc[c * 4 + 2] += v0.z + v1.z;
            acc[c * 4 + 3] += v0.w + v1.w;
        }
    }
    if (e < d) {
        int s0 = __builtin_amdgcn_readfirstlane(col[start + e]);
        const float* xr0 = x + (size_t)s0 * DD;
#pragma unroll
        for (int c = 0; c < DCH; ++c) {
            float4 v0 = *(const float4*)(xr0 + c * 128 + lane * 4);
            acc[c * 4 + 0] += v0.x;
            acc[c * 4 + 1] += v0.y;
            acc[c * 4 + 2] += v0.z;
            acc[c * 4 + 3] += v0.w;
        }
    }
    const float inv = 1.f / (float)(d > 0 ? d : 1);
    float* orow = out + (size_t)node * DD;
#pragma unroll
    for (int c = 0; c < DCH; ++c) {
        float4 v;
        v.x = acc[c * 4 + 0] * inv;
        v.y = acc[c * 4 + 1] * inv;
        v.z = acc[c * 4 + 2] * inv;
        v.w = acc[c * 4 + 3] * inv;
        if (relu) {
            v.x = fmaxf(v.x, 0.f); v.y = fmaxf(v.y, 0.f);
            v.z = fmaxf(v.z, 0.f); v.w = fmaxf(v.w, 0.f);
        }
        *(float4*)(orow + c * 128 + lane * 4) = v;
    }
}

// ---------------------------------------------------------------------------
// Fused SAGE GEMM:  C = A0 @ W0 + A1 @ W1 + bias  (optional ReLU)
// N fixed = 128 cols. K per stream is runtime (multiple of 128).
// Block: 256 thr = 8 waves; each wave owns a 16x128 strip (8 WMMA 16x16 tiles).
// Weight slab staged quad-interleaved in 64KB LDS:
//   lds[q*512 + n*4 + j] = W[kh + 4q + j][n]
// -> each B fragment {W[k][n], W[k+1][n]} is one contiguous ds_load_b64,
//    and the two half-waves hit disjoint bank sets (4n,4n+1 vs 4n+2,4n+3).
// Uses V_WMMA_F32_16X16X4_F32 (wave32 fp32 matrix pipe).
// ---------------------------------------------------------------------------
__global__ void __launch_bounds__(256) k_sage_gemm(const float* __restrict__ A0,
                                                   const float* __restrict__ A1,
                                                   const float* __restrict__ W0,
                                                   const float* __restrict__ W1,
                                                   const float* __restrict__ bias,
                                                   float* __restrict__ Cout,
                                                   int M, int K, int relu) {
    extern __shared__ float ldsW[];  // 128 * 128 floats = 64KB
    const int tid = threadIdx.x;
    const int wave = tid >> 5;
    const int lane = tid & 31;
    const int hf = lane >> 4;    // half-wave select
    const int l15 = lane & 15;
    const int rowBase = blockIdx.x * 128 + wave * 16;
    const int arow_i = rowBase + l15;  // A-fragment row for this lane (M = lane&15)
    // Clamp instead of predicating: WMMA row m of A only feeds output row m,
    // and OOB output rows are store-guarded, so clamped garbage is harmless.
    const int arow_c = arow_i < M ? arow_i : (M - 1);

    v8f c[8];
#pragma unroll
    for (int t = 0; t < 8; ++t) c[t] = (v8f)(0.f);

    for (int s = 0; s < 2; ++s) {
        const float* __restrict__ A = s ? A1 : A0;
        const float* __restrict__ W = s ? W1 : W0;
        const float* arow = A + (size_t)arow_c * K;
        for (int kh = 0; kh < K; kh += 128) {
            __syncthreads();
            // cooperative quad-interleaved stage of a 128x128 fp32 weight slab
            for (int i = tid; i < 32 * 128; i += 256) {
                const int q = i >> 7;       // k-quad 0..31
                const int n = i & 127;      // column
                const float* wr = W + (size_t)(kh + 4 * q) * 128 + n;
                float4 wv;
                wv.x = wr[0];
                wv.y = wr[128];
                wv.z = wr[256];
                wv.w = wr[384];
                *(float4*)(ldsW + q * 512 + n * 4) = wv;
            }
            __syncthreads();
#pragma unroll 2
            for (int k0 = 0; k0 < 128; k0 += 4) {
                // A fragment: VGPR0 = A[m][k+2h], VGPR1 = A[m][k+2h+1]
                const float2 av = *(const float2*)(arow + kh + k0 + 2 * hf);
                const v2f a = (v2f){av.x, av.y};
                // B fragments: one ds_load_b64 each, batched before the WMMAs
                const float* bbase = ldsW + (k0 >> 2) * 512 + l15 * 4 + 2 * hf;
                v2f bfrag[8];
#pragma unroll
                for (int nt = 0; nt < 8; ++nt)
                    bfrag[nt] = *(const v2f*)(bbase + nt * 64);
#pragma unroll
                for (int nt = 0; nt < 8; ++nt)
                    c[nt] = __builtin_amdgcn_wmma_f32_16x16x4_f32(
                        false, a, false, bfrag[nt], (short)0, c[nt], false, false);
            }
        }
    }

    // epilogue: C/D layout — VGPR v: lanes0-15 -> M=v, lanes16-31 -> M=v+8
#pragma unroll
    for (int nt = 0; nt < 8; ++nt) {
        const int coln = nt * 16 + l15;
        const float bv = bias[coln];
#pragma unroll
        for (int v = 0; v < 8; ++v) {
            const int r = rowBase + v + 8 * hf;
            if (r < M) {
                float val = c[nt][v] + bv;
                if (relu) val = fmaxf(val, 0.f);
                Cout[(size_t)r * 128 + coln] = val;
            }
        }
    }
}

// ---------------------------------------------------------------------------
// x0c[i] = [ x1[down_dst[i]] , x0[i] ]   (up-scatter over identity down_src)
// ---------------------------------------------------------------------------
__global__ void __launch_bounds__(256) k_cat_up(const float* __restrict__ x1,
                                                const float* __restrict__ x0,
                                                const int* __restrict__ down_dst,
                                                float* __restrict__ x0c, int n0) {
    const int wave = threadIdx.x >> 5;
    const int lane = threadIdx.x & 31;
    const int row = blockIdx.x * 8 + wave;
    if (row >= n0) return;
    const int dd = __builtin_amdgcn_readfirstlane(down_dst[row]);
    float4 v1 = *(const float4*)(x1 + (size_t)dd * 128 + lane * 4);
    float4 v0 = *(const float4*)(x0 + (size_t)row * 128 + lane * 4);
    *(float4*)(x0c + (size_t)row * 256 + lane * 4) = v1;
    *(float4*)(x0c + (size_t)row * 256 + 128 + lane * 4) = v0;
}

// ---------------------------------------------------------------------------
// Readout: logits (wave dot) + softmax over all nodes
// ---------------------------------------------------------------------------
__global__ void __launch_bounds__(256) k_logits(const float* __restrict__ x0f,
                                                const float* __restrict__ Wm,
                                                const float* __restrict__ bm,
                                                float* __restrict__ logits,
                                                float* __restrict__ pmax, int n) {
    __shared__ float smax[8];
    const int wave = threadIdx.x >> 5;
    const int lane = threadIdx.x & 31;
    const int node = blockIdx.x * 8 + wave;
    float partial = 0.f;
    if (node < n) {
        float4 xv = *(const float4*)(x0f + (size_t)node * 128 + lane * 4);
        float4 wv = *(const float4*)(Wm + lane * 4);
        partial = xv.x * wv.x + xv.y * wv.y + xv.z * wv.z + xv.w * wv.w;
    }
#pragma unroll
    for (int off = 16; off; off >>= 1) partial += __shfl_xor(partial, off, 32);
    const float l = partial + bm[0];
    if (node < n && lane == 0) logits[node] = l;
    if (lane == 0) smax[wave] = (node < n) ? l : -FLT_MAX;
    __syncthreads();
    if (threadIdx.x == 0) {
        float m = smax[0];
#pragma unroll
        for (int i = 1; i < 8; ++i) m = fmaxf(m, smax[i]);
        pmax[blockIdx.x] = m;
    }
}

__global__ void __launch_bounds__(256) k_reduce_max(const float* __restrict__ p,
                                                    int n, float* __restrict__ g) {
    __shared__ float s[256];
    const int t = threadIdx.x;
    float m = -FLT_MAX;
    for (int i = t; i < n; i += 256) m = fmaxf(m, p[i]);
    s[t] = m;
    __syncthreads();
    for (int off = 128; off; off >>= 1) {
        if (t < off) s[t] = fmaxf(s[t], s[t + off]);
        __syncthreads();
    }
    if (t == 0) *g = s[0];
}

__global__ void __launch_bounds__(256) k_exp(float* __restrict__ logits,
                                             const float* __restrict__ gmax,
                                             float* __restrict__ psum, int n) {
    __shared__ float s[256];
    const int t = threadIdx.x;
    const int i = blockIdx.x * 256 + t;
    float e = 0.f;
    if (i < n) {
        e = __expf(logits[i] - *gmax);
        logits[i] = e;
    }
    s[t] = e;
    __syncthreads();
    for (int off = 128; off; off >>= 1) {
        if (t < off) s[t] += s[t + off];
        __syncthreads();
    }
    if (t == 0) psum[blockIdx.x] = s[0];
}

__global__ void __launch_bounds__(256) k_reduce_sum(const float* __restrict__ p,
                                                    int n, float* __restrict__ g) {
    __shared__ float s[256];
    const int t = threadIdx.x;
    float m = 0.f;
    for (int i = t; i < n; i += 256) m += p[i];
    s[t] = m;
    __syncthreads();
    for (int off = 128; off; off >>= 1) {
        if (t < off) s[t] += s[t + off];
        __syncthreads();
    }
    if (t == 0) *g = s[0];
}

__global__ void __launch_bounds__(256) k_div(const float* __restrict__ e,
                                             const float* __restrict__ gsum,
                                             float* __restrict__ out, int n) {
    const int i = blockIdx.x * 256 + threadIdx.x;
    if (i < n) out[i] = e[i] * (1.f / *gsum);
}

// ---------------------------------------------------------------------------
extern "C" void kernel_launch(void* const* d_in, const int* in_sizes, int n_in,
                              void* d_out, int out_size, void* d_ws, size_t ws_size,
                              hipStream_t stream) {
    constexpr int N0 = 50000, N1 = 5000, E0 = 800000, E1 = 80000;

    const float* X   = (const float*)d_in[0];
    const float* W0s = (const float*)d_in[1];
    const float* W0n = (const float*)d_in[2];
    const float* b0  = (const float*)d_in[3];
    const float* W1s = (const float*)d_in[4];
    const float* W1n = (const float*)d_in[5];
    const float* b1  = (const float*)d_in[6];
    const float* W2s = (const float*)d_in[7];
    const float* W2n = (const float*)d_in[8];
    const float* b2  = (const float*)d_in[9];
    const float* Wm  = (const float*)d_in[10];
    const float* bm  = (const float*)d_in[11];
    const int* src0 = (const int*)d_in[12];
    const int* dst0 = (const int*)d_in[13];
    const int* src1 = (const int*)d_in[14];
    const int* dst1 = (const int*)d_in[15];
    const int* dsrc = (const int*)d_in[16];
    const int* ddst = (const int*)d_in[17];
    float* out = (float*)d_out;

    // --- workspace arena ---
    char* w = (char*)d_ws;
    size_t off = 0;
    auto alloc = [&](size_t bytes) -> void* {
        off = (off + 255) & ~(size_t)255;
        void* p = w + off;
        off += bytes;
        return p;
    };
    float* f_hn0  = (float*)alloc((size_t)N0 * 128 * 4);  // hn0, later reused as x0f
    float* f_hn2  = (float*)alloc((size_t)N0 * 256 * 4);  // hn2; first half doubles as x0
    float* f_x0   = f_hn2;                                // x0 dead before hn2 is written
    float* f_x0c  = (float*)alloc((size_t)N0 * 256 * 4);
    float* f_x1a  = (float*)alloc((size_t)N1 * 128 * 4);
    float* f_hn1  = (float*)alloc((size_t)N1 * 128 * 4);
    float* f_x1   = (float*)alloc((size_t)N1 * 128 * 4);
    float* f_log  = (float*)alloc((size_t)N0 * 4);
    const int NBL = (N0 + 7) / 8;                 // logits blocks
    const int NBE = (N0 + 255) / 256;             // exp blocks
    float* f_pmax = (float*)alloc((size_t)NBL * 4);
    float* f_psum = (float*)alloc((size_t)NBE * 4);
    float* f_gmax = (float*)alloc(4);
    float* f_gsum = (float*)alloc(4);
    int* i_deg0 = (int*)alloc((size_t)N0 * 4);
    int* i_rp0  = (int*)alloc((size_t)(N0 + 1) * 4);
    int* i_cur0 = (int*)alloc((size_t)N0 * 4);
    int* i_col0 = (int*)alloc((size_t)E0 * 4);
    int* i_deg1 = (int*)alloc((size_t)N1 * 4);
    int* i_rp1  = (int*)alloc((size_t)(N1 + 1) * 4);
    int* i_cur1 = (int*)alloc((size_t)N1 * 4);
    int* i_col1 = (int*)alloc((size_t)E1 * 4);
    int* i_degd = (int*)alloc((size_t)N1 * 4);
    int* i_rpd  = (int*)alloc((size_t)(N1 + 1) * 4);
    int* i_curd = (int*)alloc((size_t)N1 * 4);
    int* i_cold = (int*)alloc((size_t)N0 * 4);
    (void)ws_size; (void)n_in; (void)in_sizes; (void)out_size;

    // --- CSR build for the three edge sets ---
    hipMemsetAsync(i_deg0, 0, (size_t)N0 * 4, stream);
    hipMemsetAsync(i_deg1, 0, (size_t)N1 * 4, stream);
    hipMemsetAsync(i_degd, 0, (size_t)N1 * 4, stream);
    k_count_deg<<<(E0 + 255) / 256, 256, 0, stream>>>(dst0, i_deg0, E0);
    k_count_deg<<<(E1 + 255) / 256, 256, 0, stream>>>(dst1, i_deg1, E1);
    k_count_deg<<<(N0 + 255) / 256, 256, 0, stream>>>(ddst, i_degd, N0);
    k_scan<<<1, 1024, 0, stream>>>(i_deg0, i_rp0, i_cur0, N0);
    k_scan<<<1, 1024, 0, stream>>>(i_deg1, i_rp1, i_cur1, N1);
    k_scan<<<1, 1024, 0, stream>>>(i_degd, i_rpd, i_curd, N1);
    k_fill_col<<<(E0 + 255) / 256, 256, 0, stream>>>(src0, dst0, i_cur0, i_col0, E0);
    k_fill_col<<<(E1 + 255) / 256, 256, 0, stream>>>(src1, dst1, i_cur1, i_col1, E1);
    k_fill_col<<<(N0 + 255) / 256, 256, 0, stream>>>(dsrc, ddst, i_curd, i_cold, N0);

    const size_t ldsBytes = 128 * 128 * sizeof(float);  // 64KB weight slab

    // --- SAGE layer 0 (no relu): x0 = X@W0s + mean_nbr(X)@W0n + b0 ---
    k_gather_mean<1><<<(N0 + 7) / 8, 256, 0, stream>>>(X, i_rp0, i_deg0, i_col0,
                                                       f_hn0, N0, 0);
    k_sage_gemm<<<(N0 + 127) / 128, 256, ldsBytes, stream>>>(X, f_hn0, W0s, W0n, b0,
                                                             f_x0, N0, 128, 0);

    // --- down pool: x1a = relu(mean over clusters of x0) ---
    k_gather_mean<1><<<(N1 + 7) / 8, 256, 0, stream>>>(f_x0, i_rpd, i_degd, i_cold,
                                                       f_x1a, N1, 1);

    // --- SAGE layer 1 (+relu) on lv1 graph ---
    k_gather_mean<1><<<(N1 + 7) / 8, 256, 0, stream>>>(f_x1a, i_rp1, i_deg1, i_col1,
                                                       f_hn1, N1, 0);
    k_sage_gemm<<<(N1 + 127) / 128, 256, ldsBytes, stream>>>(f_x1a, f_hn1, W1s, W1n,
                                                             b1, f_x1, N1, 128, 1);

    // --- up sweep: x0c = [x1[down_dst], x0]  (down_src == arange) ---
    k_cat_up<<<(N0 + 7) / 8, 256, 0, stream>>>(f_x1, f_x0, ddst, f_x0c, N0);

    // --- SAGE layer 2 (+relu), K=256 per stream; x0 is dead, hn2 overlays it ---
    k_gather_mean<2><<<(N0 + 7) / 8, 256, 0, stream>>>(f_x0c, i_rp0, i_deg0, i_col0,
                                                       f_hn2, N0, 0);
    float* f_x0f = f_hn0;  // hn0 dead -> reuse as x0f
    k_sage_gemm<<<(N0 + 127) / 128, 256, ldsBytes, stream>>>(f_x0c, f_hn2, W2s, W2n,
                                                             b2, f_x0f, N0, 256, 1);

    // --- readout: Linear(128,1) + softmax over nodes ---
    k_logits<<<NBL, 256, 0, stream>>>(f_x0f, Wm, bm, f_log, f_pmax, N0);
    k_reduce_max<<<1, 256, 0, stream>>>(f_pmax, NBL, f_gmax);
    k_exp<<<NBE, 256, 0, stream>>>(f_log, f_gmax, f_psum, N0);
    k_reduce_sum<<<1, 256, 0, stream>>>(f_psum, NBE, f_gsum);
    k_div<<<NBE, 256, 0, stream>>>(f_log, f_gsum, out, N0);
}